// LSTM_37778532335896
// MI455X (gfx1250) — compile-verified
//
#include <hip/hip_runtime.h>
#include <hip/hip_bf16.h>

typedef float v2f __attribute__((ext_vector_type(2)));
typedef float v8f __attribute__((ext_vector_type(8)));

#define HIDDEN 10
#define T_LEN  2048
#define BATCH  4096
#define INPUT  4
#define PF     8      // x prefetch depth (one 128B cacheline = 8 timesteps)

// gfx1250 V_TANH_F32 (TRANS pipe) -- confirmed lowering in round 4.
#if __has_builtin(__builtin_amdgcn_tanhf)
  __device__ __forceinline__ float dev_tanh(float x) { return __builtin_amdgcn_tanhf(x); }
#elif __has_builtin(__builtin_amdgcn_tanh_f32)
  __device__ __forceinline__ float dev_tanh(float x) { return __builtin_amdgcn_tanh_f32(x); }
#else
  __device__ __forceinline__ float dev_tanh(float x) {
    float r = __builtin_amdgcn_rcpf(__expf(2.0f * x) + 1.0f);
    return __builtin_fmaf(-2.0f, r, 1.0f);
  }
#endif

__device__ __forceinline__ float fast_sigmoid(float x) {
    // sigmoid(x) = 0.5*tanh(x/2) + 0.5  -> 1 TRANS + 2 VALU
    return __builtin_fmaf(0.5f, dev_tanh(0.5f * x), 0.5f);
}
__device__ __forceinline__ float fast_tanh(float x) { return dev_tanh(x); }

// Scheduling-only barrier: same-wave DS ops execute in-order (ISA), so a
// single-wave workgroup needs NO memory fence for its LDS round trip --
// avoids __syncthreads()'s s_wait_storecnt/dscnt fence on the critical path.
__device__ __forceinline__ void wave_order_barrier() {
    __builtin_amdgcn_wave_barrier();
}

// One 32-thread (single-wave) block handles a 16-row batch tile for all T steps.
// WMMA f32 16x16x4 layout (wave32):
//   A (16x4):  lane L, vgpr v -> A[M = L%16][K = 2*(L/16) + v]
//   B (4x16):  lane L, vgpr v -> B[K = 2*(L/16) + v][N = L%16]
//   C/D (16x16): lane L, vgpr v -> C[M = v + 8*(L/16)][N = L%16]
__global__ __launch_bounds__(32)
void lstm_wmma_f32_kernel(const float* __restrict__ x,
                          const float* __restrict__ W_ih,
                          const float* __restrict__ W_hh,
                          const float* __restrict__ b_ih,
                          const float* __restrict__ b_hh,
                          const float* __restrict__ W_lin,
                          const float* __restrict__ b_lin,
                          float* __restrict__ out)
{
    // h tile, COLUMN-major: hlds2[col n][row r] -> contiguous 8-row spans per lane
    __shared__ __align__(32) float hlds2[16 * 16];

    const int lane = threadIdx.x;     // 0..31
    const int m    = lane & 15;       // A-row (batch row within tile); also C/D column n
    const int half = lane >> 4;       // 0/1: which K/M half this lane covers
    const int n    = m;               // C/D column = hidden/gate index
    const int b0   = blockIdx.x * 16; // first batch row of this tile

    // ---------------- hoist weights into fragments (once) ----------------
    v2f Bih[4];        // per gate: W_ih^T padded to 4x16
    v2f Bhh[4][3];     // per gate, per K-chunk (K = 0..3,4..7,8..11)
    v8f Cbias[4];      // per gate: bias broadcast down the column
    float wl[HIDDEN];  // full W_lin (epilogue)
    float wl6[6];      // W_lin entries matching this lane's ha K-slots

    #pragma unroll
    for (int g = 0; g < 4; ++g) {
        float e0 = 0.f, e1 = 0.f, bia = 0.f;
        if (n < HIDDEN) {
            const int row = g * HIDDEN + n;        // PyTorch gate order i,f,g,o
            const int k0  = 2 * half;              // INPUT == 4 == K exactly
            e0  = W_ih[row * INPUT + k0];
            e1  = W_ih[row * INPUT + k0 + 1];
            bia = b_ih[row] + b_hh[row];
        }
        Bih[g].x = e0; Bih[g].y = e1;
        v8f c;
        #pragma unroll
        for (int e = 0; e < 8; ++e) c[e] = bia;    // bias depends only on column n
        Cbias[g] = c;

        #pragma unroll
        for (int j = 0; j < 3; ++j) {
            float h0 = 0.f, h1 = 0.f;
            if (n < HIDDEN) {
                const int row = g * HIDDEN + n;
                const int kk  = j * 4 + 2 * half;
                if (kk     < HIDDEN) h0 = W_hh[row * HIDDEN + kk];
                if (kk + 1 < HIDDEN) h1 = W_hh[row * HIDDEN + kk + 1];
            }
            Bhh[g][j].x = h0; Bhh[g][j].y = h1;
        }
    }
    #pragma unroll
    for (int i = 0; i < HIDDEN; ++i) wl[i] = W_lin[i];
    #pragma unroll
    for (int j = 0; j < 3; ++j) {
        const int kk = j * 4 + 2 * half;
        wl6[2 * j + 0] = (kk     < HIDDEN) ? W_lin[kk]     : 0.f;
        wl6[2 * j + 1] = (kk + 1 < HIDDEN) ? W_lin[kk + 1] : 0.f;
    }
    const float blin = b_lin[0];

    // zero h tile (h0 = 0); padding columns stay 0 forever by construction
    for (int i = lane; i < 256; i += 32) hlds2[i] = 0.f;
    __syncthreads();   // once, before the loop: full barrier is fine here

    v8f creg;                                  // c state: 8 batch rows, column n
    #pragma unroll
    for (int e = 0; e < 8; ++e) creg[e] = 0.f;

    const float* xrow = x + ((size_t)(b0 + m) * T_LEN) * INPUT + 2 * half;
    float*       orow = out + (size_t)(b0 + m) * T_LEN;

    // -------- software-pipelined x fetch: batch-issue PF loads per body --------
    v2f xbuf[PF], xcur[PF];
    #pragma unroll
    for (int p = 0; p < PF; ++p) {
        xbuf[p].x = xrow[(size_t)p * INPUT + 0];
        xbuf[p].y = xrow[(size_t)p * INPUT + 1];
    }

    for (int tb = 0; tb < T_LEN; tb += PF) {
        // consume previous body's loads (wait lands here with 8 steps of slack) ...
        #pragma unroll
        for (int p = 0; p < PF; ++p) xcur[p] = xbuf[p];

        // ... then put the next PF loads in flight immediately.
        // Clamp the BASE once per body: only the final body would run past T-1,
        // and its prefetch (stale but in-bounds) is never consumed.
        {
            int tnext = tb + PF;
            if (tnext > T_LEN - PF) tnext = T_LEN - PF;
            const float* xpf = xrow + (size_t)tnext * INPUT;
            #pragma unroll
            for (int p = 0; p < PF; ++p) {
                xbuf[p].x = xpf[p * INPUT + 0];   // 8 loads, one base + imm offsets
                xbuf[p].y = xpf[p * INPUT + 1];
            }
        }

        #pragma unroll
        for (int u = 0; u < PF; ++u) {
            const int t = tb + u;
            const v2f xa = xcur[u];

            // A fragments of h_{t-1}: A[m][kk+v] = hlds2[kk+v][m]
            v2f ha[3];
            #pragma unroll
            for (int j = 0; j < 3; ++j) {
                const int kk = j * 4 + 2 * half;
                ha[j].x = hlds2[(kk    ) * 16 + m];
                ha[j].y = hlds2[(kk + 1) * 16 + m];
            }

            // input projection + bias: one exact-f32 WMMA per gate
            v8f acc[4];
            #pragma unroll
            for (int g = 0; g < 4; ++g)
                acc[g] = __builtin_amdgcn_wmma_f32_16x16x4_f32(
                    false, xa, false, Bih[g], (short)0, Cbias[g], false, false);

            // recurrent h_{t-1} @ W_hh^T: 3 K-chunks of 4
            #pragma unroll
            for (int j = 0; j < 3; ++j)
                #pragma unroll
                for (int g = 0; g < 4; ++g)
                    acc[g] = __builtin_amdgcn_wmma_f32_16x16x4_f32(
                        false, ha[j], false, Bhh[g][j], (short)0, acc[g], false, false);

            // output head for t-1, reusing ha: lane pair (L, L^16) holds all 12 K-slots
            {
                float po = ha[0].x * wl6[0] + ha[0].y * wl6[1]
                         + ha[1].x * wl6[2] + ha[1].y * wl6[3]
                         + ha[2].x * wl6[4] + ha[2].y * wl6[5];
                po += __shfl_xor(po, 16, 32);
                if (half == 0) {
                    const int to = (t > 0) ? (t - 1) : 0;  // t==0 dummy, overwritten by t==1
                    orow[to] = po + blin;
                }
            }

            wave_order_barrier();   // order only: h_{t-1} loads precede the overwrite

            // cell update: identical lane layout across all 4 gate tiles -> pure VALU
            v8f hv;
            #pragma unroll
            for (int e = 0; e < 8; ++e) {
                const float iv = fast_sigmoid(acc[0][e]);
                const float fv = fast_sigmoid(acc[1][e]);
                const float gv = fast_tanh   (acc[2][e]);
                const float ov = fast_sigmoid(acc[3][e]);
                const float cv = fv * creg[e] + iv * gv;
                creg[e] = cv;
                hv[e] = ov * fast_tanh(cv);
            }
            // store h_t column-major: one contiguous 32B span -> 2x ds_store_b128;
            // same-wave DS is in-order, next iteration's loads need no fence
            *(v8f*)&hlds2[n * 16 + 8 * half] = hv;
            wave_order_barrier();
        }
    }

    // epilogue: out[b, T-1] from the final h tile (same-wave DS in-order)
    if (half == 0) {
        float o = blin;
        #pragma unroll
        for (int i = 0; i < HIDDEN; ++i) o += hlds2[i * 16 + m] * wl[i];
        orow[T_LEN - 1] = o;
    }
}

extern "C" void kernel_launch(void* const* d_in, const int* in_sizes, int n_in,
                              void* d_out, int out_size, void* d_ws, size_t ws_size,
                              hipStream_t stream) {
    const float* x     = (const float*)d_in[0];
    const float* W_ih  = (const float*)d_in[1];
    const float* W_hh  = (const float*)d_in[2];
    const float* b_ih  = (const float*)d_in[3];
    const float* b_hh  = (const float*)d_in[4];
    const float* W_lin = (const float*)d_in[5];
    const float* b_lin = (const float*)d_in[6];
    float* out = (float*)d_out;

    dim3 grid(BATCH / 16);   // 256 independent 16-row tiles
    dim3 block(32);          // one wave32 per tile
    lstm_wmma_f32_kernel<<<grid, block, 0, stream>>>(
        x, W_ih, W_hh, b_ih, b_hh, W_lin, b_lin, out);
}